// FineGrainedFeatureEditing_5394478924639
// MI455X (gfx1250) — compile-verified
//
#include <hip/hip_runtime.h>
#include <hip/hip_bf16.h>

// Sizes from the reference
#define BB    8
#define CCH   256
#define HH    32
#define WW    32
#define KC    50
#define NPIX  (HH*WW)          // 1024
#define PTOT  (BB*NPIX)        // 8192
#define NTILE (PTOT/16)        // 512 column tiles of 16
#define LDA   260              // padded LDS stride for A tile
#define SWLD  17               // padded stride for SW staging

typedef float v2f __attribute__((ext_vector_type(2)));
typedef float v8f __attribute__((ext_vector_type(8)));

// ---------------------------------------------------------------------------
// sim tile: D[16x16] = A(16x256 rows of Xn, in LDS) x B(256x16 = Xn^T cols, global)
// fp32 WMMA 16x16x4, 64 chained MACs over the 256-channel K dimension.
// A layout (ISA 7.12.2): a.x = A[M=lane&15][k0 + (lane<16?0:2)], a.y = +1
// B layout (row-striped): b.x = B[k0 + (lane<16?0:2)][N=lane&15] = Xn[col=N][k]
// ---------------------------------------------------------------------------
__device__ inline v8f sim_tile(const float* __restrict__ XnCol,
                               const float* __restrict__ As,
                               int m, int koff) {
  v8f acc = {0.f,0.f,0.f,0.f,0.f,0.f,0.f,0.f};
  const float* ap = As + m * LDA + koff;
  const float* bp = XnCol + m * CCH + koff;
  #pragma unroll 8
  for (int k0 = 0; k0 < CCH; k0 += 4) {
    v2f a, b;
    a.x = ap[k0];  a.y = ap[k0 + 1];
    b.x = bp[k0];  b.y = bp[k0 + 1];
    acc = __builtin_amdgcn_wmma_f32_16x16x4_f32(false, a, false, b,
                                                (short)0, acc, false, false);
  }
  return acc;
}

// ---------------------------------------------------------------------------
// k0: zero global counters [cnt_ano, sum_d_nor, sum_d_ano, pad]
// ---------------------------------------------------------------------------
__global__ void fgfe_k0(float* counters) {
  if (threadIdx.x < 4) counters[threadIdx.x] = 0.f;
}

// ---------------------------------------------------------------------------
// k1: one wave per pixel. min dist to 50 centers, anomaly flag, norms,
// writes X and Xn row-major [P, C], accumulates Ld stats.
// ---------------------------------------------------------------------------
__global__ __launch_bounds__(256) void fgfe_k1(
    const float* __restrict__ f, const float* __restrict__ center,
    const float* __restrict__ Tc_p,
    float* __restrict__ Xn, float* __restrict__ Xw,
    float* __restrict__ nor, float* counters) {
  const int lane = threadIdx.x & 31;
  const int wave = threadIdx.x >> 5;
  const int p  = blockIdx.x * 8 + wave;
  const int bb = p >> 10;
  const int pp = p & (NPIX - 1);

  float xv[8];
  float x2 = 0.f;
  #pragma unroll
  for (int t = 0; t < 8; ++t) {
    const int c = t * 32 + lane;
    const float v = f[((size_t)bb * CCH + c) * NPIX + pp];
    xv[t] = v;
    x2 += v * v;
  }
  #pragma unroll
  for (int msk = 1; msk < 32; msk <<= 1) x2 += __shfl_xor(x2, msk, 32);

  float d2min = 3.4e38f;
  for (int k = 0; k < KC; ++k) {
    float dot = 0.f, cc = 0.f;
    #pragma unroll
    for (int t = 0; t < 8; ++t) {
      const float cv = center[k * CCH + t * 32 + lane];
      dot += xv[t] * cv;
      cc  += cv * cv;
    }
    #pragma unroll
    for (int msk = 1; msk < 32; msk <<= 1) {
      dot += __shfl_xor(dot, msk, 32);
      cc  += __shfl_xor(cc,  msk, 32);
    }
    d2min = fminf(d2min, x2 + cc - 2.0f * dot);
  }
  const float d    = sqrtf(fmaxf(d2min, 0.f));
  const float anof = (d > Tc_p[0]) ? 1.f : 0.f;
  const float inv  = 1.0f / fmaxf(sqrtf(x2), 1e-8f);   // COS_EPS

  #pragma unroll
  for (int t = 0; t < 8; ++t) {
    const int c = t * 32 + lane;
    Xw[(size_t)p * CCH + c] = xv[t];
    Xn[(size_t)p * CCH + c] = xv[t] * inv;
  }
  nor[p] = 1.0f - anof;
  if (lane == 0) {
    atomicAdd(&counters[0], anof);
    atomicAdd(&counters[1], d * (1.f - anof));
    atomicAdd(&counters[2], d * anof);
  }
}

// ---------------------------------------------------------------------------
// k2: one block per 16-row tile; 8 waves stream disjoint 16-column tiles.
// Pass 1: Z_i, E2_i -> thr_i.  Pass 2: SW tile + Y += SW @ X via WMMA.
// ---------------------------------------------------------------------------
__global__ __launch_bounds__(256) void fgfe_k2(
    const float* __restrict__ Xn, const float* __restrict__ Xw,
    const float* __restrict__ nor, const float* __restrict__ counters,
    const float* __restrict__ memry, const int* __restrict__ is_obj_p,
    float* __restrict__ out) {
  __shared__ float As[16 * LDA];              // Xn row tile
  __shared__ float swBuf[8][16 * SWLD];       // per-wave SW staging
  __shared__ float Yblk[16 * CCH];            // cross-wave Y accumulation
  __shared__ float zPart[8][16], e2Part[8][16];
  __shared__ float invZ[16], thr[16], Sblk[16];

  const int tid     = threadIdx.x;
  const int lane    = tid & 31;
  const int wave    = tid >> 5;
  const int rowBase = blockIdx.x * 16;
  const int half    = lane >> 4;        // 0 or 1
  const int m       = lane & 15;
  const int koff    = half * 2;

  for (int i = tid; i < 16 * CCH; i += 256) {
    const int r = i >> 8, k = i & 255;
    As[r * LDA + k] = Xn[(size_t)(rowBase + r) * CCH + k];
    Yblk[i] = 0.f;
  }
  if (tid < 16) Sblk[tid] = 0.f;
  __syncthreads();

  const float t_norm = fmaxf((float)PTOT - counters[0], 1.0f);
  const float mu     = 1.0f / t_norm;

  // ---------------- Pass 1: Z and E2 per row ----------------
  float zacc[8], e2acc[8];
  #pragma unroll
  for (int v = 0; v < 8; ++v) { zacc[v] = 0.f; e2acc[v] = 0.f; }

  for (int tile = wave; tile < NTILE; tile += 8) {
    const int colBase = tile * 16;
    const v8f s = sim_tile(Xn + (size_t)colBase * CCH, As, m, koff);
    const float norc = nor[colBase + m];
    #pragma unroll
    for (int v = 0; v < 8; ++v) {
      const float e = __expf(s[v]) * norc;
      zacc[v]  += e;
      e2acc[v] += e * e;
    }
  }
  #pragma unroll
  for (int v = 0; v < 8; ++v) {
    #pragma unroll
    for (int msk = 1; msk < 16; msk <<= 1) {
      zacc[v]  += __shfl_xor(zacc[v],  msk, 16);
      e2acc[v] += __shfl_xor(e2acc[v], msk, 16);
    }
  }
  if (m == 0) {
    #pragma unroll
    for (int v = 0; v < 8; ++v) {
      zPart[wave][v + half * 8]  = zacc[v];
      e2Part[wave][v + half * 8] = e2acc[v];
    }
  }
  __syncthreads();
  if (tid < 16) {
    float Z = 0.f, E2 = 0.f;
    for (int w = 0; w < 8; ++w) { Z += zPart[w][tid]; E2 += e2Part[w][tid]; }
    const float iz  = 1.0f / Z;
    const float var = fmaxf(E2 * iz * iz * mu - mu * mu, 0.0f);
    invZ[tid] = iz;
    thr[tid]  = mu - 2.0f * sqrtf(var);
  }
  __syncthreads();

  // ---------------- Pass 2: SW + Y = SW @ X ----------------
  float ssum[8];
  #pragma unroll
  for (int v = 0; v < 8; ++v) ssum[v] = 0.f;
  v8f yfrag[16];
  #pragma unroll
  for (int ch = 0; ch < 16; ++ch) yfrag[ch] = (v8f){0.f,0.f,0.f,0.f,0.f,0.f,0.f,0.f};

  float* mySw = swBuf[wave];
  for (int tile = wave; tile < NTILE; tile += 8) {   // 64 uniform iterations/wave
    const int colBase = tile * 16;
    const v8f s = sim_tile(Xn + (size_t)colBase * CCH, As, m, koff);
    const float norc = nor[colBase + m];
    #pragma unroll
    for (int v = 0; v < 8; ++v) {
      const int row = v + half * 8;
      const float w    = __expf(s[v]) * invZ[row];
      const float diff = w - thr[row];
      const float sv   = fmaxf(diff, 0.f) * w / (fabsf(diff) + 1e-10f) * norc;
      ssum[v] += sv;
      mySw[row * SWLD + m] = sv;          // C-layout -> LDS
    }
    __syncthreads();                      // order writes before cross-lane reads

    #pragma unroll
    for (int ch = 0; ch < 16; ++ch) {
      v8f y = yfrag[ch];
      #pragma unroll
      for (int k0 = 0; k0 < 16; k0 += 4) {
        v2f a, b;
        a.x = mySw[m * SWLD + k0 + koff];          // A = SW, M=m, K=col-in-tile
        a.y = mySw[m * SWLD + k0 + koff + 1];
        const float* xp = Xw + (size_t)(colBase + k0 + koff) * CCH + ch * 16 + m;
        b.x = xp[0];                               // B = X[colpix][chan]
        b.y = xp[CCH];
        y = __builtin_amdgcn_wmma_f32_16x16x4_f32(false, a, false, b,
                                                  (short)0, y, false, false);
      }
      yfrag[ch] = y;
    }
    __syncthreads();                      // reads done before next overwrite
  }

  // reduce S across lanes/waves
  #pragma unroll
  for (int v = 0; v < 8; ++v) {
    #pragma unroll
    for (int msk = 1; msk < 16; msk <<= 1) ssum[v] += __shfl_xor(ssum[v], msk, 16);
  }
  if (m == 0) {
    #pragma unroll
    for (int v = 0; v < 8; ++v) atomicAdd(&Sblk[v + half * 8], ssum[v]);
  }
  // reduce Y across waves
  #pragma unroll
  for (int ch = 0; ch < 16; ++ch) {
    #pragma unroll
    for (int v = 0; v < 8; ++v)
      atomicAdd(&Yblk[(v + half * 8) * CCH + ch * 16 + m], yfrag[ch][v]);
  }
  __syncthreads();

  // final write: f_out[b, c, h, w]
  const int isObj = *is_obj_p;
  for (int i = tid; i < 16 * CCH; i += 256) {
    const int r = i >> 8, c = i & 255;
    const int g  = rowBase + r;
    const int bb = g >> 10, pp = g & (NPIX - 1);
    const float xval = Xw[(size_t)g * CCH + c];
    float val;
    if (nor[g] != 0.0f)      val = xval;                         // normal pixel
    else if (isObj)          val = memry[c * NPIX + pp];         // is_object path
    else                     val = Yblk[r * CCH + c] / Sblk[r];  // edited feature
    out[((size_t)bb * CCH + c) * NPIX + pp] = val;
  }
}

// ---------------------------------------------------------------------------
// k3: Ld scalar
// ---------------------------------------------------------------------------
__global__ void fgfe_k3(const float* counters, float* out) {
  if (threadIdx.x == 0 && blockIdx.x == 0) {
    const float ca = counters[0];
    const float cn = (float)PTOT - ca;
    const float mean_nor = counters[1] / fmaxf(cn, 1.f);
    const float mean_ano = counters[2] / fmaxf(ca, 1.f);
    out[(size_t)BB * CCH * NPIX] = (ca > 0.f) ? mean_nor / (mean_ano + 0.001f)
                                              : mean_nor;
  }
}

extern "C" void kernel_launch(void* const* d_in, const int* in_sizes, int n_in,
                              void* d_out, int out_size, void* d_ws, size_t ws_size,
                              hipStream_t stream) {
  const float* f      = (const float*)d_in[0];   // [8,256,32,32]
  const float* center = (const float*)d_in[1];   // [50,256]
  const float* Tc     = (const float*)d_in[2];   // scalar
  const float* memry  = (const float*)d_in[3];   // [1,256,32,32]
  const int*   is_obj = (const int*)d_in[4];     // scalar int
  float* out = (float*)d_out;

  float* ws       = (float*)d_ws;
  float* Xn       = ws;                              // P*C = 2,097,152 floats
  float* Xw       = ws + (size_t)PTOT * CCH;         // P*C
  float* nor      = ws + (size_t)2 * PTOT * CCH;     // P
  float* counters = nor + PTOT;                      // 4

  fgfe_k0<<<1, 64, 0, stream>>>(counters);
  fgfe_k1<<<PTOT / 8, 256, 0, stream>>>(f, center, Tc, Xn, Xw, nor, counters);
  fgfe_k2<<<NTILE, 256, 0, stream>>>(Xn, Xw, nor, counters, memry, is_obj, out);
  fgfe_k3<<<1, 1, 0, stream>>>(counters, out);
}